// HGT_85993835200828
// MI455X (gfx1250) — compile-verified
//
#include <hip/hip_runtime.h>
#include <math.h>

#define NN       20000
#define EE       250000
#define HEADS    2
#define DHEAD    64
#define NEG_SLOPE 0.01f
#define RT       5            // row tiles per block in proj128 (80 rows)

typedef float v2f __attribute__((ext_vector_type(2)));
typedef float v8f __attribute__((ext_vector_type(8)));

__device__ __forceinline__ float leakyf(float x) { return x > 0.f ? x : NEG_SLOPE * x; }

__device__ __forceinline__ v8f wmma4(v2f a, v2f b, v8f c) {
  return __builtin_amdgcn_wmma_f32_16x16x4_f32(false, a, false, b, (short)0,
                                               c, false, false);
}

// ---------------------------------------------------------------------------
// Feature encoder. One block = 16 nodes.
//   waves 0-1 : des   (K=768) -> h0 cols   0..31   (WMMA, unguarded)
//   waves 2-3 : tweet (K=768) -> h0 cols  32..63   (WMMA, unguarded)
//   waves 4-5 : num   (K=5)   -> h0 cols  64..95   (scalar VALU)
//   waves 6-7 : cat   (K=11)  -> h0 cols  96..127  (scalar VALU)
// then feat = leaky(h0 @ w_lin + b_lin)  (WMMA, A from LDS)
// feat written into d_out columns 128..255 (stride 256).
// ---------------------------------------------------------------------------
__global__ __launch_bounds__(256) void encoder_kernel(
    const float* __restrict__ x,
    const float* __restrict__ w_des,  const float* __restrict__ b_des,
    const float* __restrict__ w_tw,   const float* __restrict__ b_tw,
    const float* __restrict__ w_num,  const float* __restrict__ b_num,
    const float* __restrict__ w_cat,  const float* __restrict__ b_cat,
    const float* __restrict__ w_lin,  const float* __restrict__ b_lin,
    float* __restrict__ out)          // N x 256, cols 128..255
{
  __shared__ float h0[16 * 128];      // 8 KB
  const int row0  = blockIdx.x * 16;
  const int wave  = threadIdx.x >> 5;
  const int lane  = threadIdx.x & 31;
  const int r     = lane & 15;
  const int koff  = (lane >> 4) << 1;
  const int n     = lane & 15;
  const int mbase = (lane >> 4) << 3;

  // x row layout: [num(5) | tweet(768) | cat(11) | des(768)]
  if (wave < 4) {
    // ---- big segments via WMMA, K = 768 (literal, no guards) ----
    const int isDes = (wave < 2);
    const float* Wp  = isDes ? w_des : w_tw;
    const float* bp  = isDes ? b_des : b_tw;
    const int    xo  = isDes ? 784 : 5;
    const int    hb  = isDes ? 0 : 32;
    const int    wcol = (wave & 1) * 16;

    const float* Ap = x + (size_t)(row0 + r) * 1552 + xo + koff;
    const float* Bp = Wp + wcol + r;               // ldb = 32

    v8f acc = {0.f, 0.f, 0.f, 0.f, 0.f, 0.f, 0.f, 0.f};
#pragma unroll 4
    for (int k = 0; k < 768; k += 4) {
      v2f a, b;
      a.x = Ap[k];
      a.y = Ap[k + 1];
      b.x = Bp[(k + koff) * 32];
      b.y = Bp[(k + koff + 1) * 32];
      acc = wmma4(a, b, acc);
    }
    const float bv = bp[wcol + n];
#pragma unroll
    for (int j = 0; j < 8; ++j)
      h0[(mbase + j) * 128 + hb + wcol + n] = leakyf(acc[j] + bv);
  } else {
    // ---- tiny segments via plain VALU dot products ----
    const int isNum = (wave < 6);
    const float* Wp = isNum ? w_num : w_cat;
    const float* bp = isNum ? b_num : b_cat;
    const int    xo = isNum ? 0 : 773;
    const int    hb = isNum ? 64 : 96;
    const int    K  = isNum ? 5 : 11;
    // wave pair covers 16 rows x 32 cols; even wave of pair -> even rows
    const int sub = (wave - 4) & 1;                // 0 or 1 within pair
#pragma unroll
    for (int i = 0; i < 8; ++i) {
      const int row = 2 * i + sub;                 // 0..15
      const int col = lane;                        // 0..31
      const float* xr = x + (size_t)(row0 + row) * 1552 + xo;
      float s = bp[col];
      for (int k = 0; k < K; ++k)
        s += xr[k] * Wp[k * 32 + col];
      h0[row * 128 + hb + col] = leakyf(s);
    }
  }
  __syncthreads();

  // feat = leaky(h0 @ w_lin + b_lin); A from LDS (indexed -> ds_load)
  const int col0 = wave * 16;
  v8f acc2 = {0.f, 0.f, 0.f, 0.f, 0.f, 0.f, 0.f, 0.f};
  const float* Bl = w_lin + col0 + r;
#pragma unroll 4
  for (int k = 0; k < 128; k += 4) {
    const int k0 = k + koff;
    v2f a, b;
    a.x = h0[r * 128 + k0];
    a.y = h0[r * 128 + k0 + 1];
    b.x = Bl[k0 * 128];
    b.y = Bl[(k0 + 1) * 128];
    acc2 = wmma4(a, b, acc2);
  }
  {
    const float bv = b_lin[col0 + n];
#pragma unroll
    for (int j = 0; j < 8; ++j) {
      const int row = row0 + mbase + j;
      out[(size_t)row * 256 + 128 + col0 + n] = leakyf(acc2[j] + bv);
    }
  }
}

// ---------------------------------------------------------------------------
// (N x 128) @ (128 x 128) + bias.  Block = 80 rows (5 tiles) x 128 cols.
// A slab staged in LDS (40 KB); each B register pair reused by 5 WMMAs.
// ---------------------------------------------------------------------------
__global__ __launch_bounds__(256) void proj128_kernel(
    const float* __restrict__ X, int ldx,
    const float* __restrict__ W, const float* __restrict__ Bc,
    float* __restrict__ OUT)
{
  __shared__ float As[RT * 16 * 128];           // 40 KB
  const int row0 = blockIdx.x * (RT * 16);

  for (int i = threadIdx.x; i < RT * 16 * 32; i += 256) {
    const int rr = i >> 5, c4 = i & 31;
    ((float4*)As)[i] = *(const float4*)(X + (size_t)(row0 + rr) * ldx + c4 * 4);
  }
  __syncthreads();

  const int wave  = threadIdx.x >> 5;
  const int lane  = threadIdx.x & 31;
  const int r     = lane & 15;
  const int koff  = (lane >> 4) << 1;
  const int n     = lane & 15;
  const int mbase = (lane >> 4) << 3;
  const int col0  = wave * 16;

  v8f acc[RT];
#pragma unroll
  for (int t = 0; t < RT; ++t) acc[t] = (v8f){0.f,0.f,0.f,0.f,0.f,0.f,0.f,0.f};

  const float* Bp = W + col0 + r;
#pragma unroll 2
  for (int k = 0; k < 128; k += 4) {
    const int k0 = k + koff;
    v2f b;
    b.x = Bp[k0 * 128];
    b.y = Bp[(k0 + 1) * 128];
#pragma unroll
    for (int t = 0; t < RT; ++t) {
      v2f a;
      a.x = As[(t * 16 + r) * 128 + k0];
      a.y = As[(t * 16 + r) * 128 + k0 + 1];
      acc[t] = wmma4(a, b, acc[t]);
    }
  }

  const float bv = Bc[col0 + n];
#pragma unroll
  for (int t = 0; t < RT; ++t)
#pragma unroll
    for (int j = 0; j < 8; ++j)
      OUT[(size_t)(row0 + t * 16 + mbase + j) * 128 + col0 + n] = acc[t][j] + bv;
}

// ---------------------------------------------------------------------------
// Fuse relation matrices into projection weights:
//   W_ke[e][c][h*64+el] = sum_d wk[c][h*64+d] * a_rel[e][h][d][el]  (same for v)
// ---------------------------------------------------------------------------
__global__ void fuse_w_kernel(const float* __restrict__ wk, const float* __restrict__ bk,
                              const float* __restrict__ wv, const float* __restrict__ bv,
                              const float* __restrict__ a_rel, const float* __restrict__ m_rel,
                              float* __restrict__ Wke, float* __restrict__ bke,
                              float* __restrict__ Wve, float* __restrict__ bve)
{
  const int t = blockIdx.x * blockDim.x + threadIdx.x;
  const int total = 2 * 2 * 128 * 128;  // kv, e, c, j
  if (t < total) {
    const int j  = t & 127;
    const int c  = (t >> 7) & 127;
    const int e  = (t >> 14) & 1;
    const int kv = t >> 15;
    const int h  = j >> 6, el = j & 63;
    const float* w     = kv ? wv : wk;
    const float* relhe = (kv ? m_rel : a_rel) + (size_t)(e * HEADS + h) * DHEAD * DHEAD;
    float s = 0.f;
    for (int d = 0; d < DHEAD; ++d)
      s += w[c * 128 + h * 64 + d] * relhe[d * 64 + el];
    (kv ? Wve : Wke)[(size_t)(e * 128 + c) * 128 + j] = s;
  }
  const int tb = t - total;
  if (tb >= 0 && tb < 2 * 2 * 128) {
    const int j  = tb & 127;
    const int e  = (tb >> 7) & 1;
    const int kv = tb >> 8;
    const int h  = j >> 6, el = j & 63;
    const float* bsrc  = kv ? bv : bk;
    const float* relhe = (kv ? m_rel : a_rel) + (size_t)(e * HEADS + h) * DHEAD * DHEAD;
    float s = 0.f;
    for (int d = 0; d < DHEAD; ++d)
      s += bsrc[h * 64 + d] * relhe[d * 64 + el];
    (kv ? bve : bke)[e * 128 + j] = s;
  }
}

// ---------------------------------------------------------------------------
// Attention passes (segment softmax over dst, per edge type)
// ---------------------------------------------------------------------------
__device__ __forceinline__ void atomicMaxF(float* addr, float val) {
  if (val >= 0.f) atomicMax((int*)addr, __float_as_int(val));
  else            atomicMin((unsigned int*)addr, __float_as_uint(val));
}

__global__ void init_msg_kernel(float* __restrict__ msg, int n) {
  const int i = blockIdx.x * blockDim.x + threadIdx.x;
  if (i < n) msg[i] = 0.f;
}
__global__ void init_maxsum_kernel(float* __restrict__ mx, float* __restrict__ sm, int n) {
  const int i = blockIdx.x * blockDim.x + threadIdx.x;
  if (i < n) { mx[i] = -INFINITY; sm[i] = 0.f; }
}

__global__ void score_kernel(const float* __restrict__ q, const float* __restrict__ ke,
                             const int* __restrict__ src, const int* __restrict__ dst,
                             const float* __restrict__ p_rel,
                             float* __restrict__ raw, float* __restrict__ mx)
{
  const int t = blockIdx.x * blockDim.x + threadIdx.x;
  if (t >= EE * HEADS) return;
  const int e = t >> 1, h = t & 1;
  const int s = src[e], d = dst[e];
  const float4* qa = (const float4*)(q  + (size_t)d * 128 + h * 64);
  const float4* ka = (const float4*)(ke + (size_t)s * 128 + h * 64);
  float acc = 0.f;
#pragma unroll 4
  for (int i = 0; i < 16; ++i) {
    const float4 qv = qa[i], kv = ka[i];
    acc += qv.x * kv.x + qv.y * kv.y + qv.z * kv.z + qv.w * kv.w;
  }
  acc *= p_rel[h] * 0.125f;  // / sqrt(64)
  raw[t] = acc;
  atomicMaxF(&mx[d * HEADS + h], acc);
}

__global__ void expsum_kernel(const int* __restrict__ dst, const float* __restrict__ mx,
                              float* __restrict__ raw, float* __restrict__ sm)
{
  const int t = blockIdx.x * blockDim.x + threadIdx.x;
  if (t >= EE * HEADS) return;
  const int e = t >> 1, h = t & 1;
  const int d = dst[e];
  const float ex = expf(raw[t] - mx[d * HEADS + h]);
  raw[t] = ex;
  atomicAdd(&sm[d * HEADS + h], ex);
}

__global__ void scatter_kernel(const float* __restrict__ ve,
                               const int* __restrict__ src, const int* __restrict__ dst,
                               const float* __restrict__ raw, const float* __restrict__ sm,
                               float* __restrict__ msg)
{
  const int t = blockIdx.x * blockDim.x + threadIdx.x;
  if (t >= EE * HEADS) return;
  const int e = t >> 1, h = t & 1;
  const int s = src[e], d = dst[e];
  const float coef = raw[t] / (sm[d * HEADS + h] + 1e-16f);
  const float* vv = ve  + (size_t)s * 128 + h * 64;
  float*       mm = msg + (size_t)d * 128 + h * 64;
#pragma unroll 4
  for (int i = 0; i < 64; ++i) atomicAdd(&mm[i], vv[i] * coef);
}

__global__ void gelu_kernel(const float* __restrict__ in, float* __restrict__ out, int n) {
  const int i = blockIdx.x * blockDim.x + threadIdx.x;
  if (i < n) {
    const float v = in[i];
    out[i] = 0.5f * v * (1.f + erff(v * 0.70710678118654752f));
  }
}

// h_out = leaky( beta*t2 + (1-beta)*Xin ), beta = sigmoid(skip)
__global__ void blend_kernel(const float* __restrict__ t2,
                             const float* __restrict__ Xin, int ldx,
                             const float* __restrict__ skip,
                             float* __restrict__ OUT, int ldo)
{
  const int i = blockIdx.x * blockDim.x + threadIdx.x;
  if (i >= NN * 128) return;
  const int row = i >> 7, c = i & 127;
  const float beta = 1.f / (1.f + expf(-skip[0]));
  const float v = beta * t2[i] + (1.f - beta) * Xin[(size_t)row * ldx + c];
  OUT[(size_t)row * ldo + c] = leakyf(v);
}

// ---------------------------------------------------------------------------
extern "C" void kernel_launch(void* const* d_in, const int* in_sizes, int n_in,
                              void* d_out, int out_size, void* d_ws, size_t ws_size,
                              hipStream_t stream) {
  (void)in_sizes; (void)n_in; (void)out_size; (void)ws_size;

  const float* x      = (const float*)d_in[0];
  const int*   ei[2]  = {(const int*)d_in[1], (const int*)d_in[2]};
  const float* w_des  = (const float*)d_in[3];  const float* b_des = (const float*)d_in[4];
  const float* w_tw   = (const float*)d_in[5];  const float* b_tw  = (const float*)d_in[6];
  const float* w_num  = (const float*)d_in[7];  const float* b_num = (const float*)d_in[8];
  const float* w_cat  = (const float*)d_in[9];  const float* b_cat = (const float*)d_in[10];
  const float* w_lin  = (const float*)d_in[11]; const float* b_lin = (const float*)d_in[12];

  struct HGT { const float *wk,*bk,*wq,*bq,*wv,*bv,*wa,*ba,*a_rel,*m_rel,*p_rel,*skip; };
  HGT hg[2];
  for (int L = 0; L < 2; ++L) {
    const int b = 13 + L * 12;
    hg[L].wk    = (const float*)d_in[b + 0];  hg[L].bk   = (const float*)d_in[b + 1];
    hg[L].wq    = (const float*)d_in[b + 2];  hg[L].bq   = (const float*)d_in[b + 3];
    hg[L].wv    = (const float*)d_in[b + 4];  hg[L].bv   = (const float*)d_in[b + 5];
    hg[L].wa    = (const float*)d_in[b + 6];  hg[L].ba   = (const float*)d_in[b + 7];
    hg[L].a_rel = (const float*)d_in[b + 8];  hg[L].m_rel= (const float*)d_in[b + 9];
    hg[L].p_rel = (const float*)d_in[b + 10]; hg[L].skip = (const float*)d_in[b + 11];
  }

  float* out = (float*)d_out;  // N x 256 : [h2 | feat]

  // ---- workspace carve-up (floats) ----
  float* ws = (float*)d_ws;
  size_t off = 0;
  float* hbuf = ws + off; off += (size_t)NN * 128;       // layer-1 output h1
  float* qb   = ws + off; off += (size_t)NN * 128;       // q  (reused: gelu out)
  float* keb  = ws + off; off += (size_t)2 * NN * 128;   // ke[2] (reused: wa out)
  float* veb  = ws + off; off += (size_t)2 * NN * 128;   // ve[2]
  float* msg  = ws + off; off += (size_t)NN * 128;       // aggregated messages
  float* raw  = ws + off; off += (size_t)EE * HEADS;     // scores -> exp
  float* mx   = ws + off; off += (size_t)NN * HEADS;
  float* sm   = ws + off; off += (size_t)NN * HEADS;
  float* Wke  = ws + off; off += 2 * 128 * 128;
  float* bke  = ws + off; off += 2 * 128;
  float* Wve  = ws + off; off += 2 * 128 * 128;
  float* bve  = ws + off; off += 2 * 128;

  const int TB = 256;
  const int encBlocks  = NN / 16;                      // 1250
  const int projBlocks = NN / (RT * 16);               // 250
  const int fuseGrid   = (2 * 2 * 128 * 128 + 2 * 2 * 128 + TB - 1) / TB;
  const int edgeGrid   = (EE * HEADS + TB - 1) / TB;
  const int nodeGrid   = (NN * 128 + TB - 1) / TB;
  const int msGrid     = (NN * HEADS + TB - 1) / TB;

  // ---- encoder: feat -> out[:,128:256] ----
  encoder_kernel<<<encBlocks, TB, 0, stream>>>(
      x, w_des, b_des, w_tw, b_tw, w_num, b_num, w_cat, b_cat, w_lin, b_lin, out);

  for (int L = 0; L < 2; ++L) {
    const HGT& p = hg[L];
    const float* X  = (L == 0) ? (out + 128) : hbuf;    // layer input
    const int    ldx = (L == 0) ? 256 : 128;

    fuse_w_kernel<<<fuseGrid, TB, 0, stream>>>(
        p.wk, p.bk, p.wv, p.bv, p.a_rel, p.m_rel, Wke, bke, Wve, bve);

    proj128_kernel<<<projBlocks, TB, 0, stream>>>(X, ldx, p.wq, p.bq, qb);
    for (int e = 0; e < 2; ++e) {
      proj128_kernel<<<projBlocks, TB, 0, stream>>>(
          X, ldx, Wke + (size_t)e * 128 * 128, bke + e * 128,
          keb + (size_t)e * NN * 128);
      proj128_kernel<<<projBlocks, TB, 0, stream>>>(
          X, ldx, Wve + (size_t)e * 128 * 128, bve + e * 128,
          veb + (size_t)e * NN * 128);
    }

    init_msg_kernel<<<nodeGrid, TB, 0, stream>>>(msg, NN * 128);
    for (int e = 0; e < 2; ++e) {
      const int* src = ei[e];
      const int* dst = ei[e] + EE;
      init_maxsum_kernel<<<msGrid, TB, 0, stream>>>(mx, sm, NN * HEADS);
      score_kernel<<<edgeGrid, TB, 0, stream>>>(
          qb, keb + (size_t)e * NN * 128, src, dst, p.p_rel + e * HEADS, raw, mx);
      expsum_kernel<<<edgeGrid, TB, 0, stream>>>(dst, mx, raw, sm);
      scatter_kernel<<<edgeGrid, TB, 0, stream>>>(
          veb + (size_t)e * NN * 128, src, dst, raw, sm, msg);
    }

    // out_proj = gelu(msg) @ wa + ba ; blend with skip; leaky
    gelu_kernel<<<nodeGrid, TB, 0, stream>>>(msg, qb, NN * 128);       // qb = gelu
    proj128_kernel<<<projBlocks, TB, 0, stream>>>(qb, 128, p.wa, p.ba, keb);
    if (L == 0)
      blend_kernel<<<nodeGrid, TB, 0, stream>>>(keb, X, ldx, p.skip, hbuf, 128);
    else
      blend_kernel<<<nodeGrid, TB, 0, stream>>>(keb, X, ldx, p.skip, out, 256);
  }
}